// FastWeightAttention_1297080123584
// MI455X (gfx1250) — compile-verified
//
#include <hip/hip_runtime.h>
#include <hip/hip_bf16.h>

typedef float v2f __attribute__((ext_vector_type(2)));
typedef float v4f __attribute__((ext_vector_type(4)));
typedef float v8f __attribute__((ext_vector_type(8)));

#define B_   1024
#define D_   1024
#define H_   16
#define DK_  64
#define PHI_ 128
#define EPS_ 1e-6f

// GEMM tiling
#define BM   128
#define BN   64
#define BK   32
#define APAD 4   // As row stride 36 floats: b64 even-aligned, v4f 16B-aligned
#define WPAD 2   // Wt row stride 34 floats: b64 even-aligned, banks spread

// ---------------------------------------------------------------------------
// GEMM: C[M,N] = A[M,K] * W[K,N] (+ bias[N]), f32 via V_WMMA_F32_16X16X4_F32.
// Block = 256 threads = 8 waves; block tile 128x64; wave tile 16x64.
// LDS double-buffered: A as [BM][BK+4] row-major (K-pair contiguous ->
// ds_load_b64), W transposed into [BN][BK+2] (K-pair contiguous per column).
// Global side is all coalesced b128. No divergence -> EXEC all ones (WMMA req).
// ---------------------------------------------------------------------------
__global__ __launch_bounds__(256)
void fwa_gemm_f32_wmma(const float* __restrict__ A, const float* __restrict__ W,
                       float* __restrict__ C, const float* __restrict__ bias,
                       int M, int N, int K) {
  __shared__ float As[2][BM][BK + APAD];
  __shared__ float Wt[2][BN][BK + WPAD];

  const int tid   = threadIdx.x;
  const int lane  = tid & 31;
  const int wave  = tid >> 5;
  const int m0    = blockIdx.x * BM;
  const int n0    = blockIdx.y * BN;
  const int mloc  = wave * 16 + (lane & 15);   // wave's row within block tile
  const int khalf = (lane >> 4) << 1;          // 0 or 2: K-pair select
  const int nloc  = lane & 15;

  v8f acc[4];
  acc[0] = v8f{}; acc[1] = v8f{}; acc[2] = v8f{}; acc[3] = v8f{};

  auto load_tile = [&](int kt, int buf) {
    const int k0 = kt * BK;
    // A tile: 128x32 f32 = 1024 v4f, 4 per thread (coalesced b128)
#pragma unroll
    for (int i = 0; i < 4; ++i) {
      const int f  = tid + i * 256;
      const int m  = f >> 3;             // 0..127
      const int kq = (f & 7) << 2;       // 0,4,..,28
      v4f val = *(const v4f*)(A + (size_t)(m0 + m) * K + k0 + kq);
      *(v4f*)(&As[buf][m][kq]) = val;
    }
    // W tile: 32x64 f32 = 512 v4f, 2 per thread (coalesced b128), transposed
#pragma unroll
    for (int i = 0; i < 2; ++i) {
      const int f  = tid + i * 256;
      const int r  = f >> 4;             // 0..31 (K row)
      const int nq = (f & 15) << 2;      // 0,4,..,60
      v4f val = *(const v4f*)(W + (size_t)(k0 + r) * N + n0 + nq);
      Wt[buf][nq + 0][r] = val.x;
      Wt[buf][nq + 1][r] = val.y;
      Wt[buf][nq + 2][r] = val.z;
      Wt[buf][nq + 3][r] = val.w;
    }
  };

  load_tile(0, 0);
  const int nk = K / BK;
  for (int kt = 0; kt < nk; ++kt) {
    __syncthreads();                       // tile[cur] ready; prev reads done
    const int cur = kt & 1;
    if (kt + 1 < nk) load_tile(kt + 1, cur ^ 1);   // overlap with WMMA below
#pragma unroll
    for (int ks = 0; ks < BK; ks += 4) {
      const int ka = ks + khalf;
      v2f a  = *(const v2f*)(&As[cur][mloc][ka]);            // ds_load_b64
      v2f b0 = *(const v2f*)(&Wt[cur][nloc +  0][ka]);
      v2f b1 = *(const v2f*)(&Wt[cur][nloc + 16][ka]);
      v2f b2 = *(const v2f*)(&Wt[cur][nloc + 32][ka]);
      v2f b3 = *(const v2f*)(&Wt[cur][nloc + 48][ka]);
      acc[0] = __builtin_amdgcn_wmma_f32_16x16x4_f32(false, a, false, b0, (short)0, acc[0], false, false);
      acc[1] = __builtin_amdgcn_wmma_f32_16x16x4_f32(false, a, false, b1, (short)0, acc[1], false, false);
      acc[2] = __builtin_amdgcn_wmma_f32_16x16x4_f32(false, a, false, b2, (short)0, acc[2], false, false);
      acc[3] = __builtin_amdgcn_wmma_f32_16x16x4_f32(false, a, false, b3, (short)0, acc[3], false, false);
    }
  }

  // C/D layout: VGPR i -> row m0' + i + 8*(lane>>4), col = lane%16
  const int mbase = m0 + wave * 16 + ((lane >> 4) << 3);
#pragma unroll
  for (int t = 0; t < 4; ++t) {
    const int n  = n0 + t * 16 + (lane & 15);
    const float bv = bias ? bias[n] : 0.0f;
#pragma unroll
    for (int i = 0; i < 8; ++i) {
      C[(size_t)(mbase + i) * N + n] = acc[t][i] + bv;
    }
  }
}

// ---------------------------------------------------------------------------
// Gate: beta[b,h] = sigmoid(dot(x[b,:], Wg[:,h])). One wave32 per output.
// ---------------------------------------------------------------------------
__global__ __launch_bounds__(256)
void fwa_gate(const float* __restrict__ x, const float* __restrict__ Wg,
              float* __restrict__ beta) {
  const int lane = threadIdx.x & 31;
  const int idx  = blockIdx.x * (blockDim.x >> 5) + (threadIdx.x >> 5);
  if (idx >= B_ * H_) return;
  const int b = idx / H_, h = idx - b * H_;
  float s = 0.0f;
  for (int i = lane; i < D_; i += 32) s += x[(size_t)b * D_ + i] * Wg[(size_t)i * H_ + h];
#pragma unroll
  for (int o = 16; o; o >>= 1) s += __shfl_xor(s, o, 32);
  if (lane == 0) beta[idx] = 1.0f / (1.0f + __expf(-s));
}

// ---------------------------------------------------------------------------
// Fast-weight core: one block (128 thr = 4 waves) per (b,h).
//   dpfp(q), dpfp(k) in LDS; then stream weights[64,128]:
//   v_exist = W k ; delta = beta*(v - v_exist) ; w_new = W + delta*k^T ;
//   out_h   = w_new q.
// Streamed once (1 GB total) -> non-temporal b128 loads/stores so the L2
// stays reserved for the GEMM operands; butterfly shfl_xor reductions per row.
// ---------------------------------------------------------------------------
__global__ __launch_bounds__(128)
void fwa_update(const float* __restrict__ qp, const float* __restrict__ kp,
                const float* __restrict__ vp, const float* __restrict__ beta,
                const float* __restrict__ weights,
                float* __restrict__ w_new, float* __restrict__ out_h) {
  __shared__ float s_qraw[DK_], s_kraw[DK_];
  __shared__ float s_qf[PHI_], s_kf[PHI_];
  __shared__ float s_rq[4], s_rk[4];

  const int bh   = blockIdx.x;
  const int b    = bh / H_, h = bh - b * H_;
  const int t    = threadIdx.x;
  const int lane = t & 31, wave = t >> 5;
  const size_t projoff = (size_t)b * D_ + (size_t)h * DK_;

  if (t < DK_) { s_qraw[t] = qp[projoff + t]; s_kraw[t] = kp[projoff + t]; }
  __syncthreads();

  // dpfp, NU=1: x2 = [relu(x), relu(-x)] (len 128); y[i] = x2[i]*x2[(i-1)&127]
  auto x2 = [](const float* raw, int i) {
    i &= (PHI_ - 1);
    return (i < DK_) ? fmaxf(raw[i], 0.0f) : fmaxf(-raw[i - DK_], 0.0f);
  };
  const float yq = x2(s_qraw, t) * x2(s_qraw, t - 1);
  const float yk = x2(s_kraw, t) * x2(s_kraw, t - 1);

  float sq = yq, sk = yk;
#pragma unroll
  for (int o = 16; o; o >>= 1) { sq += __shfl_xor(sq, o, 32); sk += __shfl_xor(sk, o, 32); }
  if (lane == 0) { s_rq[wave] = sq; s_rk[wave] = sk; }
  __syncthreads();
  const float sumq = s_rq[0] + s_rq[1] + s_rq[2] + s_rq[3];
  const float sumk = s_rk[0] + s_rk[1] + s_rk[2] + s_rk[3];
  s_qf[t] = yq / (sumq + EPS_);
  s_kf[t] = yk / (sumk + EPS_);
  __syncthreads();

  const float bgate = beta[bh];
  const v4f kf = *(const v4f*)(s_kf + lane * 4);
  const v4f qf = *(const v4f*)(s_qf + lane * 4);
  const float* Wbh  = weights + (size_t)bh * DK_ * PHI_;
  float*       WNbh = w_new   + (size_t)bh * DK_ * PHI_;

  for (int v = wave; v < DK_; v += 4) {
    if (v + 4 < DK_)  // hint the streaming pattern (global_prefetch_b8)
      __builtin_prefetch(Wbh + (size_t)(v + 4) * PHI_ + lane * 4, 0, 1);
    const v4f w = __builtin_nontemporal_load(
        (const v4f*)(Wbh + (size_t)v * PHI_ + lane * 4));
    float pe = w.x * kf.x + w.y * kf.y + w.z * kf.z + w.w * kf.w;
#pragma unroll
    for (int o = 16; o; o >>= 1) pe += __shfl_xor(pe, o, 32);   // all lanes
    const float delta = bgate * (vp[projoff + v] - pe);
    v4f wn;
    wn.x = w.x + delta * kf.x;
    wn.y = w.y + delta * kf.y;
    wn.z = w.z + delta * kf.z;
    wn.w = w.w + delta * kf.w;
    __builtin_nontemporal_store(wn, (v4f*)(WNbh + (size_t)v * PHI_ + lane * 4));
    float po = wn.x * qf.x + wn.y * qf.y + wn.z * qf.z + wn.w * qf.w;
#pragma unroll
    for (int o = 16; o; o >>= 1) po += __shfl_xor(po, o, 32);
    if (lane == 0) out_h[projoff + v] = po;
  }
}

// ---------------------------------------------------------------------------
extern "C" void kernel_launch(void* const* d_in, const int* in_sizes, int n_in,
                              void* d_out, int out_size, void* d_ws, size_t ws_size,
                              hipStream_t stream) {
  const float* x       = (const float*)d_in[0];
  const float* weights = (const float*)d_in[1];
  const float* Wq      = (const float*)d_in[2];
  const float* Wk      = (const float*)d_in[3];
  const float* Wv      = (const float*)d_in[4];
  const float* Wg      = (const float*)d_in[5];
  const float* Wo      = (const float*)d_in[6];
  const float* bo      = (const float*)d_in[7];

  float* out   = (float*)d_out;                       // [B, D]
  float* w_new = out + (size_t)B_ * D_;               // [B, H, DK, PHI]

  const size_t BD = (size_t)B_ * D_;
  float* qp    = (float*)d_ws;
  float* kp    = qp + BD;
  float* vpbuf = kp + BD;
  float* outh  = vpbuf + BD;
  float* betab = outh + BD;                           // B*H floats

  dim3 gblk(256), ggrid(B_ / BM, D_ / BN);
  // Projections: q, k, v
  fwa_gemm_f32_wmma<<<ggrid, gblk, 0, stream>>>(x, Wq, qp,    nullptr, B_, D_, D_);
  fwa_gemm_f32_wmma<<<ggrid, gblk, 0, stream>>>(x, Wk, kp,    nullptr, B_, D_, D_);
  fwa_gemm_f32_wmma<<<ggrid, gblk, 0, stream>>>(x, Wv, vpbuf, nullptr, B_, D_, D_);
  // Gate
  fwa_gate<<<(B_ * H_ + 7) / 8, 256, 0, stream>>>(x, Wg, betab);
  // Fast-weight read / update / write  (memory-bound: ~1 GB NT stream)
  fwa_update<<<B_ * H_, 128, 0, stream>>>(qp, kp, vpbuf, betab, weights, w_new, outh);
  // Output projection with bias
  fwa_gemm_f32_wmma<<<ggrid, gblk, 0, stream>>>(outh, Wo, out, bo, B_, D_, D_);
}